// Part_Attention_86346022519404
// MI455X (gfx1250) — compile-verified
//
#include <hip/hip_runtime.h>
#include <hip/hip_bf16.h>
#include <stdint.h>

typedef float v2f __attribute__((ext_vector_type(2)));
typedef float v8f __attribute__((ext_vector_type(8)));

#define LL 12
#define BB 16
#define VV 4
#define HH 12
#define TT 196
#define SS 197
#define BV (BB*VV)                    // 64
#define GLAYER (BB*HH*25)             // 4800 floats per layer of global attn
#define MAT (SS*SS)                   // 38809
#define VLAYER ((size_t)BV*HH*MAT)    // floats per layer of views attn
#define CH 16                         // rows staged per async chunk

// ---------------------------------------------------------------------------
// Kernel 1: global attention chain via WMMA. One wave per (b,h).
// 5x5 matrices zero-padded to 16x16: product of padded = padded product, so
// the top-left 5x5 block of the chained result is exact. Chain is
// carry = M_k x carry, computed as 2 x V_WMMA_F32_16X16X4_F32 (K=0..7; padded
// A columns >=5 are zero so K=8..15 contribute nothing).
// ---------------------------------------------------------------------------
__global__ __launch_bounds__(32)
void global_chain_wmma(const float* __restrict__ glob, float* __restrict__ gvals) {
    __shared__ float carry[256];                 // 16x16 running product
    const int lane = threadIdx.x;
    const int bh   = blockIdx.x;                 // b*HH + h
    const float* g0 = glob + (size_t)bh * 25;

    for (int idx = lane; idx < 256; idx += 32) { // carry = pad(M_0)
        int r = idx >> 4, c = idx & 15;
        carry[idx] = (r < 5 && c < 5) ? g0[r * 5 + c] : 0.0f;
    }
    __syncthreads();

    const int mrow = lane & 15;                  // A: row M, B/D: column N
    const int kb   = (lane < 16) ? 0 : 2;        // K sub-slot per half-wave

    for (int k = 1; k < LL; ++k) {
        const float* gk = glob + (size_t)k * GLAYER + (size_t)bh * 25;
        v8f acc = {};
        #pragma unroll
        for (int c4 = 0; c4 < 8; c4 += 4) {      // two K=4 chunks (K = 0..7)
            const int ka = c4 + kb;
            v2f A;                               // A operand: pad(M_k)[mrow][ka], [ka+1]
            A.x = (mrow < 5 && ka     < 5) ? gk[mrow * 5 + ka    ] : 0.0f;
            A.y = (mrow < 5 && ka + 1 < 5) ? gk[mrow * 5 + ka + 1] : 0.0f;
            v2f Bv;                              // B operand: carry[ka][n], carry[ka+1][n]
            Bv.x = carry[(ka    ) * 16 + mrow];
            Bv.y = carry[(ka + 1) * 16 + mrow];
            acc = __builtin_amdgcn_wmma_f32_16x16x4_f32(
                false, A, false, Bv, (short)0, acc, false, false);
        }
        __syncthreads();
        const int row0 = (lane < 16) ? 0 : 8;    // D layout -> LDS row-major
        #pragma unroll
        for (int r = 0; r < 8; ++r) carry[(row0 + r) * 16 + mrow] = acc[r];
        __syncthreads();
    }
    // row 0, cols 1..4 of the chained product
    if (lane < VV) gvals[(size_t)bh * VV + lane] = carry[1 + lane];
}

// ---------------------------------------------------------------------------
// Kernel 2: head-max -> softmax -> token budgets (RNE rounding = jnp.round).
// ---------------------------------------------------------------------------
__global__ __launch_bounds__(32)
void counts_kernel(const float* __restrict__ gvals, float* __restrict__ outCounts,
                   int* __restrict__ countsI) {
    const int b = threadIdx.x;
    if (b >= BB) return;
    float gv[VV];
    for (int v = 0; v < VV; ++v) {
        float m = -1e30f;
        for (int h = 0; h < HH; ++h)
            m = fmaxf(m, gvals[(size_t)(b * HH + h) * VV + v]);
        gv[v] = m;
    }
    float mx = fmaxf(fmaxf(gv[0], gv[1]), fmaxf(gv[2], gv[3]));
    float e[VV], s = 0.0f;
    for (int v = 0; v < VV; ++v) { e[v] = expf(gv[v] - mx); s += e[v]; }
    int cnt[VV], tot = 0;
    for (int v = 0; v < VV; ++v) {
        cnt[v] = (int)rintf(e[v] / s * (float)TT);
        tot += cnt[v];
    }
    cnt[0] += TT - tot;                          // balance to exactly TT tokens
    for (int v = 0; v < VV; ++v) {
        outCounts[b * VV + v] = (float)cnt[v];
        countsI [b * VV + v] = cnt[v];
    }
}

// ---------------------------------------------------------------------------
// Kernel 3: views rollout as a vector-matrix chain (row 0 only), streaming
// each 197x197 matrix once through LDS via async global->LDS copies (GV mode:
// full 64-bit per-lane address in a VGPR pair, "off" for the absent saddr).
//   r := row0(M_{L-1});  for k = L-2..0:  r := r^T M_k
// Thread j owns column j: acc += r[i] * M[i][j] (coalesced row-major reads).
// ---------------------------------------------------------------------------
__global__ __launch_bounds__(256)
void views_rollout(const float* __restrict__ views, float* __restrict__ lastOut) {
    __shared__ float rvec[SS + 3];
    __shared__ float stage[CH * SS];             // 16 rows = 12608 B
    const int tid = threadIdx.x;
    const size_t bvh = blockIdx.x;               // (b*V+v)*H + h
    const float* base = views + bvh * (size_t)MAT;

    for (int j = tid; j < SS; j += 256)          // r = row 0 of layer L-1
        rvec[j] = base[(size_t)(LL - 1) * VLAYER + j];
    __syncthreads();

    const int NCOPY = (CH * SS + 255) / 256;     // 13 async issues per thread

    for (int k = LL - 2; k >= 0; --k) {
        const float* Mk = base + (size_t)k * VLAYER;
        float acc = 0.0f;
        for (int i0 = 0; i0 < SS; i0 += CH) {
            const int rows  = (SS - i0 < CH) ? (SS - i0) : CH;
            const int nelem = rows * SS;         // contiguous span of `rows` rows
            // Async stage chunk into LDS. Index clamping keeps the per-wave
            // issue count uniform (duplicate writes carry identical data).
            #pragma unroll
            for (int it = 0; it < NCOPY; ++it) {
                int t  = tid + it * 256;
                int tt = (t < nelem) ? t : (nelem - 1);
                uint32_t ldsa  = (uint32_t)(uint64_t)(uintptr_t)&stage[tt];
                uint64_t gaddr = (uint64_t)(uintptr_t)(Mk + (size_t)(i0 * SS + tt));
                asm volatile("global_load_async_to_lds_b32 %0, %1, off"
                             :: "v"(ldsa), "v"(gaddr) : "memory");
            }
            asm volatile("s_wait_asynccnt 0" ::: "memory");
            __syncthreads();                     // all waves' data visible
            if (tid < SS) {
                #pragma unroll 4
                for (int r = 0; r < rows; ++r)
                    acc = fmaf(rvec[i0 + r], stage[r * SS + tid], acc);
            }
            __syncthreads();                     // done reading before reuse
        }
        if (tid < SS) rvec[tid] = acc;           // r <- r^T M_k
        __syncthreads();
    }
    for (int t = tid; t < TT; t += 256)          // CLS -> patch scores
        lastOut[bvh * TT + t] = rvec[1 + t];
}

// ---------------------------------------------------------------------------
// Kernel 4: head-max -> last_map; stable ascending rank (== jnp.argsort);
// emit selected token ids (-1 padding) per the count mask.
// ---------------------------------------------------------------------------
__global__ __launch_bounds__(256)
void topk_kernel(const float* __restrict__ last, const int* __restrict__ countsI,
                 float* __restrict__ outTopk, float* __restrict__ outLast) {
    __shared__ float lm[TT];
    __shared__ int   ord[TT];
    const int bv = blockIdx.x;                   // b*V + v
    const int t  = threadIdx.x;
    if (t < TT) {
        float m = -1e30f;
        for (int h = 0; h < HH; ++h)
            m = fmaxf(m, last[(size_t)(bv * HH + h) * TT + t]);
        lm[t] = m;
        outLast[bv * TT + t] = m;
    }
    __syncthreads();
    if (t < TT) {
        float x = lm[t];
        int r = 0;
        for (int u = 0; u < TT; ++u) {
            float y = lm[u];
            r += (int)((y < x) | ((y == x) & (u < t)));   // stable rank
        }
        ord[r] = t;                              // order[rank] = index
    }
    __syncthreads();
    if (t < TT) {
        int cnt = countsI[bv];
        outTopk[bv * TT + t] = (float)((t >= TT - cnt) ? ord[t] : -1);
    }
}

// ---------------------------------------------------------------------------
extern "C" void kernel_launch(void* const* d_in, const int* in_sizes, int n_in,
                              void* d_out, int out_size, void* d_ws, size_t ws_size,
                              hipStream_t stream) {
    const float* views = (const float*)d_in[0];  // [12,64,12,197,197] f32
    const float* glob  = (const float*)d_in[1];  // [12,16,12,5,5] f32

    float* out       = (float*)d_out;
    float* outTopk   = out;                      // [16,4,196]
    float* outCounts = out + BB * VV * TT;       // [16,4]
    float* outLast   = outCounts + BB * VV;      // [16,4,196]

    float* wsGvals   = (float*)d_ws;                     // [16*12,4]
    int*   wsCountsI = (int*)(wsGvals + BB * HH * VV);   // [64]
    float* wsLast    = (float*)(wsCountsI + BB * VV);    // [64,12,196]

    hipLaunchKernelGGL(global_chain_wmma, dim3(BB * HH), dim3(32), 0, stream,
                       glob, wsGvals);
    hipLaunchKernelGGL(counts_kernel, dim3(1), dim3(32), 0, stream,
                       wsGvals, outCounts, wsCountsI);
    hipLaunchKernelGGL(views_rollout, dim3(BV * HH), dim3(256), 0, stream,
                       views, wsLast);
    hipLaunchKernelGGL(topk_kernel, dim3(BV), dim3(256), 0, stream,
                       wsLast, wsCountsI, outTopk, outLast);
}